// MAR_onesided_missingness_18408229830683
// MI455X (gfx1250) — compile-verified
//
#include <hip/hip_runtime.h>
#include <hip/hip_bf16.h>
#include <stdint.h>

// Problem dims (match reference)
#define NN 1024
#define TT 128
#define HH 128
#define N_CELLS (NN * TT)

// Launch shape
#define WAVES_PER_BLOCK 8
#define CELLS_PER_WAVE  8
#define BLOCK_THREADS   (WAVES_PER_BLOCK * 32)

// Per-cell staged data: W_in 256 f32 + b_in 128 f32 + W_out 128 f32 = 512 f32 = 2 KB
#define BUF_FLOATS 512

// The gfx1250 builtin signature (per compiler diagnostic) takes
//   (v4i addrspace(1)*, v4i addrspace(3)*, imm int offset, imm int cpol)
// where clang prints address_space(1) as `__device__`.
typedef int v4i __attribute__((vector_size(16)));
typedef __attribute__((address_space(1))) v4i as1_v4i;
typedef __attribute__((address_space(3))) v4i as3_v4i;
// Generic->AS casts via integer round-trip (low 32 bits of a generic LDS
// pointer are the LDS byte offset; generic global pointer == global address).
#define TO_AS1(p) ((as1_v4i*)(uintptr_t)(p))
#define TO_AS3(p) ((as3_v4i*)(uint32_t)(uintptr_t)(p))

#if __has_builtin(__builtin_amdgcn_global_load_async_to_lds_b128)
#define ASYNC_B128(g, l) \
    __builtin_amdgcn_global_load_async_to_lds_b128(TO_AS1(g), TO_AS3(l), 0, 0)
#else
#define ASYNC_B128(g, l)                                                 \
    asm volatile("global_load_async_to_lds_b128 %0, %1, off"             \
                 :: "v"((uint32_t)(uintptr_t)(l)),                       \
                    "v"((uint64_t)(uintptr_t)(g)) : "memory")
#endif

#if __has_builtin(__builtin_amdgcn_s_wait_asynccnt)
#define WAIT_ASYNC(n) __builtin_amdgcn_s_wait_asynccnt(n)
#else
#define WAIT_ASYNC(n) asm volatile("s_wait_asynccnt %0" :: "i"(n) : "memory")
#endif

// WAR guard: all prior DS reads must have landed in VGPRs before the async
// unit is allowed to overwrite that LDS buffer.
#define WAIT_DS0() asm volatile("s_wait_dscnt 0x0" ::: "memory")

__global__ __launch_bounds__(BLOCK_THREADS)
void mar_cell_mlp_kernel(const float* __restrict__ X,
                         const float* __restrict__ W_in,
                         const float* __restrict__ b_in,
                         const float* __restrict__ W_out,
                         const float* __restrict__ b_out,
                         const int*   __restrict__ col_idx,
                         float* __restrict__ out)
{
    // 8 waves * 2 buffers * 2KB = 32 KB LDS
    __shared__ __align__(16) float lds[WAVES_PER_BLOCK * 2 * BUF_FLOATS];

    const int lane  = threadIdx.x & 31;
    const int wave  = threadIdx.x >> 5;
    const int cell0 = (blockIdx.x * WAVES_PER_BLOCK + wave) * CELLS_PER_WAVE;

    float* wbuf = &lds[wave * 2 * BUF_FLOATS];

    // Stage one cell's parameters (2 KB) into LDS buffer `bufsel`.
    // LDS layout per buffer: [0,256) W_in | [256,384) b_in | [384,512) W_out
    auto stage_cell = [&](int cell, int bufsel) {
        const float* g0 = W_in  + (size_t)cell * (2 * HH) + lane * 4; // W_in lo
        const float* g1 = g0 + 128;                                   // W_in hi
        const float* g2 = b_in  + (size_t)cell * HH + lane * 4;
        const float* g3 = W_out + (size_t)cell * HH + lane * 4;
        float* l0 = wbuf + bufsel * BUF_FLOATS + lane * 4;
        ASYNC_B128(g0, l0);
        ASYNC_B128(g1, l0 + 128);
        ASYNC_B128(g2, l0 + 256);
        ASYNC_B128(g3, l0 + 384);
    };

    // Prologue: stage first cell into buffer 0.
    stage_cell(cell0, 0);

#pragma unroll
    for (int i = 0; i < CELLS_PER_WAVE; ++i) {
        const int cell   = cell0 + i;
        const int bufsel = i & 1;

        if (i + 1 < CELLS_PER_WAVE) {
            // Before overwriting the other buffer, make sure the DS reads that
            // consumed it (previous iteration) have completed.
            WAIT_DS0();
            stage_cell(cell0 + i + 1, (i + 1) & 1);
            WAIT_ASYNC(4);   // first 4 async loads (this cell's buffer) done
        } else {
            WAIT_ASYNC(0);   // drain
        }

        // Uniform per-cell scalars.
        const int   t       = cell & (TT - 1);
        const int   rowbase = cell - t;            // n * T
        const int   c       = col_idx[cell];
        const float xc      = X[rowbase + c];      // neighbor column value
        const float xt      = X[cell];             // own value
        const float bo      = b_out[cell];

        // Lane l handles hidden units 4l..4l+3.
        const float* bp = wbuf + bufsel * BUF_FLOATS;
        const float4 wa = *(const float4*)(bp + lane * 8);        // h=4l,4l+1 (k pairs)
        const float4 wb = *(const float4*)(bp + lane * 8 + 4);    // h=4l+2,4l+3
        const float4 bi = *(const float4*)(bp + 256 + lane * 4);
        const float4 wo = *(const float4*)(bp + 384 + lane * 4);

        const float h0 = fmaf(wa.x, xc, fmaf(wa.y, xt, bi.x));
        const float h1 = fmaf(wa.z, xc, fmaf(wa.w, xt, bi.y));
        const float h2 = fmaf(wb.x, xc, fmaf(wb.y, xt, bi.z));
        const float h3 = fmaf(wb.z, xc, fmaf(wb.w, xt, bi.w));

        float p = wo.x * fmaxf(h0, 0.0f);
        p = fmaf(wo.y, fmaxf(h1, 0.0f), p);
        p = fmaf(wo.z, fmaxf(h2, 0.0f), p);
        p = fmaf(wo.w, fmaxf(h3, 0.0f), p);

        // wave32 xor-shuffle reduction over 32 lanes
        for (int m = 16; m > 0; m >>= 1)
            p += __shfl_xor(p, m, 32);

        if (lane == 0) {
            const float logit = p + bo;
            out[cell] = 1.0f / (1.0f + __expf(-logit));
        }
    }
}

extern "C" void kernel_launch(void* const* d_in, const int* in_sizes, int n_in,
                              void* d_out, int out_size, void* d_ws, size_t ws_size,
                              hipStream_t stream) {
    const float* X       = (const float*)d_in[0];
    const float* W_in    = (const float*)d_in[1];
    const float* b_in    = (const float*)d_in[2];
    const float* W_out   = (const float*)d_in[3];
    const float* b_out   = (const float*)d_in[4];
    const int*   col_idx = (const int*)d_in[5];
    float* out = (float*)d_out;

    const int blocks = N_CELLS / (WAVES_PER_BLOCK * CELLS_PER_WAVE); // 2048
    mar_cell_mlp_kernel<<<blocks, BLOCK_THREADS, 0, stream>>>(
        X, W_in, b_in, W_out, b_out, col_idx, out);
}